// Q_60430189855131
// MI455X (gfx1250) — compile-verified
//
#include <hip/hip_runtime.h>

typedef float v2f __attribute__((ext_vector_type(2)));
typedef float v8f __attribute__((ext_vector_type(8)));

#define HH    512
#define WW    512
#define OWD   481          // valid output rows/cols (512-32+1)
#define BANDS 31           // ceil(481/16) bands of 16 output rows
#define KROWS 48           // input rows per band (47 needed, padded to 12*4)
#define HSTR  496          // hbuf row stride (covers col reads up to 495)
#define SSTR  576          // per-wave swizzled scan-row stride (f(511)=573)
#define NCH   102
#define NBOX  1024.0f      // 32*32

// Bank-conflict-free scan-buffer swizzle: bijection of the column index.
// Writers hit addresses 18*lane + k (32 distinct banks of 64); readers span
// < 64 consecutive-ish addresses -> also conflict-free.
__device__ __forceinline__ int SW(int c) { return c + 2 * (c >> 4); }

// ---------------------------------------------------------------------------
// One workgroup = (channel, band of 16 output rows). 512 threads = 16 waves.
// Phase A: horizontal 32-box sums via prefix scans  -> hbuf[48][481] (one map)
// Phase B: vertical 32-box sums via WMMA f32 16x16x4 with banded-ones A
// Combine: per-pixel quality, masked mean partial -> partials[ch*31+band]
// ---------------------------------------------------------------------------
__global__ __launch_bounds__(512) void qilv_band_kernel(
    const float* __restrict__ O, const float* __restrict__ L,
    float* __restrict__ partials)
{
  extern __shared__ float smem[];
  float* hbuf  = smem;                 // KROWS * HSTR floats
  float* scanb = smem + KROWS * HSTR;  // 16 * SSTR floats (swizzled)

  const int band = blockIdx.x;
  const int ch   = blockIdx.y;
  const int r0   = band * 16;          // first output row of this band
  const int tid  = threadIdx.x;
  const int wave = tid >> 5;
  const int lane = tid & 31;
  float* myscan  = scanb + wave * SSTR;

  v8f acc0[5];
  v8f acc1[5];
  const bool has2 = (wave + 16) < BANDS;   // 31 column blocks over 16 waves
  const int c0A = wave * 16;
  const int c0B = c0A + 256;

  const int mi    = lane & 15;             // A-matrix M index (out row in tile)
  const int nn    = lane & 15;             // B/D N index (col in tile)
  const int khalf = (lane >> 4) << 1;      // K sub-pair select per half-wave

  // Hoisted banded-ones A-matrix values: invariant over maps and col-blocks.
  // A[m][k] = 1 iff out-row m's 32-row window covers input row k.
  float aK0[12], aK1[12];
#pragma unroll
  for (int kb = 0; kb < 12; ++kb) {
    const int g0 = (kb << 2) + khalf;
    aK0[kb] = ((unsigned)(g0 - mi)     <= 31u) ? 1.0f : 0.0f;
    aK1[kb] = ((unsigned)(g0 + 1 - mi) <= 31u) ? 1.0f : 0.0f;
  }

#pragma unroll
  for (int m = 0; m < 5; ++m) {
    // -------- Phase A: horizontal 32-box sums for map m --------
    for (int rit = 0; rit < 3; ++rit) {
      const int rr = wave + rit * 16;                 // 0..47
      const int gr = min(r0 + rr, HH - 1);            // clamp past image
      const size_t rowbase = ((size_t)ch * HH + (size_t)gr) * WW;

      float ov[16], lv[16];
      if (m != 1) {  // maps 0,2,4 need O
        const float4* p4 = reinterpret_cast<const float4*>(O + rowbase) + lane * 4;
#pragma unroll
        for (int q = 0; q < 4; ++q) {
          float4 t = p4[q];
          ov[q*4+0] = t.x; ov[q*4+1] = t.y; ov[q*4+2] = t.z; ov[q*4+3] = t.w;
        }
      }
      if (m != 0 && m != 2) {  // maps 1,3,4 need L
        const float4* p4 = reinterpret_cast<const float4*>(L + rowbase) + lane * 4;
#pragma unroll
        for (int q = 0; q < 4; ++q) {
          float4 t = p4[q];
          lv[q*4+0] = t.x; lv[q*4+1] = t.y; lv[q*4+2] = t.z; lv[q*4+3] = t.w;
        }
      }

      // lane-local inclusive prefix over its 16 contiguous columns
      float p[16];
      float run = 0.0f;
#pragma unroll
      for (int k = 0; k < 16; ++k) {
        float v;
        if      (m == 0) v = ov[k];
        else if (m == 1) v = lv[k];
        else if (m == 2) v = ov[k] * ov[k];
        else if (m == 3) v = lv[k] * lv[k];
        else             v = ov[k] * lv[k];
        run += v;
        p[k] = run;
      }
      // wave-wide exclusive scan of the 32 chunk totals
      float t = run, s = run;
#pragma unroll
      for (int d = 1; d < 32; d <<= 1) {
        float u = __shfl_up(s, d, 32);
        if (lane >= d) s += u;
      }
      const float off = s - t;
#pragma unroll
      for (int k = 0; k < 16; ++k) myscan[SW(lane * 16 + k)] = off + p[k];
      __syncthreads();
      // h[c] = inc[c+31] - inc[c-1]; stride-32 columns across lanes
#pragma unroll
      for (int kk = 0; kk < 16; ++kk) {
        const int c = lane + (kk << 5);
        if (c < OWD) {
          const float hi = myscan[SW(c + 31)];
          const float lo = (c > 0) ? myscan[SW(c - 1)] : 0.0f;
          hbuf[rr * HSTR + c] = hi - lo;
        }
      }
    }
    __syncthreads();  // hbuf(map m) complete for all 48 rows

    // -------- Phase B: vertical 32-box sum as banded-ones matmul --------
    {
      v8f a = {};
#pragma unroll
      for (int kb = 0; kb < 12; ++kb) {
        const int g0 = (kb << 2) + khalf;
        v2f av, bv;
        av.x = aK0[kb];
        av.y = aK1[kb];
        bv.x = hbuf[g0 * HSTR + c0A + nn];
        bv.y = hbuf[(g0 + 1) * HSTR + c0A + nn];
        a = __builtin_amdgcn_wmma_f32_16x16x4_f32(false, av, false, bv,
                                                  (short)0, a, false, false);
      }
      acc0[m] = a;
    }
    if (has2) {
      v8f a = {};
#pragma unroll
      for (int kb = 0; kb < 12; ++kb) {
        const int g0 = (kb << 2) + khalf;
        v2f av, bv;
        av.x = aK0[kb];
        av.y = aK1[kb];
        bv.x = hbuf[g0 * HSTR + c0B + nn];
        bv.y = hbuf[(g0 + 1) * HSTR + c0B + nn];
        a = __builtin_amdgcn_wmma_f32_16x16x4_f32(false, av, false, bv,
                                                  (short)0, a, false, false);
      }
      acc1[m] = a;
    }
    __syncthreads();  // all waves done reading hbuf before next map overwrites
  }

  // -------- Combine: exact reference conditional order --------
  auto qval = [](float so, float sl, float so2, float sl2, float sol) -> float {
    const float mul    = so * sl;
    const float sum_sq = so * so + sl * sl;
    const float num    = 4.0f * (NBOX * sol - mul) * mul;
    const float dt     = NBOX * (so2 + sl2) - sum_sq;
    const float den    = dt * sum_sq;
    float q = 1.0f;
    if (dt == 0.0f && sum_sq != 0.0f) q = 2.0f * mul / sum_sq;   // degenerate
    if (den != 0.0f)                  q = num / den;             // general
    return q;
  };

  float local = 0.0f;
  {
    const int col = c0A + nn;
#pragma unroll
    for (int v = 0; v < 8; ++v) {
      const int orow = r0 + v + ((lane >> 4) << 3);
      const float q = qval(acc0[0][v], acc0[1][v], acc0[2][v], acc0[3][v], acc0[4][v]);
      if (orow < OWD && col < OWD) local += q;
    }
  }
  if (has2) {
    const int col = c0B + nn;
#pragma unroll
    for (int v = 0; v < 8; ++v) {
      const int orow = r0 + v + ((lane >> 4) << 3);
      const float q = qval(acc1[0][v], acc1[1][v], acc1[2][v], acc1[3][v], acc1[4][v]);
      if (orow < OWD && col < OWD) local += q;
    }
  }

  // block reduction (deterministic, no atomics)
#pragma unroll
  for (int d = 16; d > 0; d >>= 1) local += __shfl_xor(local, d, 32);
  if (lane == 0) scanb[wave] = local;
  __syncthreads();
  if (tid == 0) {
    float s = 0.0f;
    for (int w = 0; w < 16; ++w) s += scanb[w];
    partials[ch * BANDS + band] = s;
  }
}

// Sum the 31 band partials per channel in a fixed order, divide by 481^2.
__global__ void qilv_finalize(const float* __restrict__ partials,
                              float* __restrict__ out)
{
  const int c = blockIdx.x * blockDim.x + threadIdx.x;
  if (c < NCH) {
    float s = 0.0f;
    for (int b = 0; b < BANDS; ++b) s += partials[c * BANDS + b];
    out[c] = s * (1.0f / (481.0f * 481.0f));
  }
}

extern "C" void kernel_launch(void* const* d_in, const int* in_sizes, int n_in,
                              void* d_out, int out_size, void* d_ws, size_t ws_size,
                              hipStream_t stream) {
  (void)in_sizes; (void)n_in; (void)out_size; (void)ws_size;
  const float* O = (const float*)d_in[0];   // "outputs"
  const float* L = (const float*)d_in[1];   // "labels"
  float* out      = (float*)d_out;          // [1,102] f32
  float* partials = (float*)d_ws;           // 102*31 f32 scratch

  const size_t shbytes = (size_t)(KROWS * HSTR + 16 * SSTR) * sizeof(float); // ~129 KB
  hipFuncSetAttribute(reinterpret_cast<const void*>(&qilv_band_kernel),
                      hipFuncAttributeMaxDynamicSharedMemorySize, (int)shbytes);

  dim3 grid(BANDS, NCH);
  qilv_band_kernel<<<grid, 512, shbytes, stream>>>(O, L, partials);
  qilv_finalize<<<(NCH + 127) / 128, 128, 0, stream>>>(partials, out);
}